// Elastic_86973087744666
// MI455X (gfx1250) — compile-verified
//
#include <hip/hip_runtime.h>

// ---------------------------------------------------------------------------
// 2-D elastic CPML FDTD (velocity-stress, 4th-order staggered) for gfx1250.
// Persistent kernel, device-wide barrier, one 16x16 tile per wave32:
//   - halo tiles staged to per-wave LDS via GLOBAL_LOAD_ASYNC_TO_LDS_B32
//   - 4th-order staggered derivatives computed as banded matmuls with
//     V_WMMA_F32_16X16X4_F32 (5 chained K=4 WMMAs per derivative)
// ---------------------------------------------------------------------------

#define NYI    256            // interior model size
#define PMLW   20
#define NYP    296            // 256 + 2*20
#define NN     (NYP * NYP)    // 87616 cells per shot
#define NTT    64             // time steps
#define NSH    2              // shots
#define NSRCS  8
#define NRECS  64

#define TILE   16
#define HB     20             // tile + 2*2 halo
#define LPAD   21             // LDS pitch (avoid bank conflicts)
#define LSZ    (HB * LPAD)    // 420 floats per halo buffer
#define TPD    19             // ceil(296/16)
#define NTILES (NSH * TPD * TPD)   // 722
#define NUMWG  64
#define BLK    256
#define WPB    8              // waves per block (wave32)
#define NWAVES (NUMWG * WPB)  // 512 persistent waves

#define DTF    5.0e-4f
#define INVDX  0.25f
#define C1F    1.125f
#define C2F    (-0.041666666666666664f)   // -1/24

// ---- workspace layout (float offsets into d_ws) ----
#define O_CNT  0                       // barrier counter (unsigned)
#define O_MAXV 4                       // max velocity (float bits, atomicMax)
#define O_BY   16
#define O_AY   (O_BY + NYP)
#define O_BX   (O_AY + NYP)
#define O_AX   (O_BX + NYP)
#define O_LAM  (O_AX + NYP)            // padded lambda [NYP*NYP]
#define O_MU   (O_LAM + NN)
#define O_BUO  (O_LAM + 2 * NN)
#define O_L2M  (O_LAM + 3 * NN)
#define O_FLD  (O_LAM + 4 * NN)        // 13 fields x NSH*NN
#define WS_FLOATS (O_FLD + 13 * NSH * NN)

typedef __attribute__((ext_vector_type(2))) float v2f;
typedef __attribute__((ext_vector_type(8))) float v8f;

#if defined(__has_builtin)
#if __has_builtin(__builtin_amdgcn_global_load_async_to_lds_b32)
#define USE_ASYNC_LDS 1
#endif
#if __has_builtin(__builtin_amdgcn_wmma_f32_16x16x4_f32)
#define USE_WMMA 1
#endif
#endif

// In HIP mode clang renders addrspace(1) as __device__, addrspace(3) as
// __shared__; the async builtin is (int AS1* src, int AS3* dst, Imm, Imm).
typedef __attribute__((address_space(1))) int g_i32_t;
typedef __attribute__((address_space(3))) int l_i32_t;

__device__ __forceinline__ void async_copy_f32(float* lds_dst, const float* g_src) {
#ifdef USE_ASYNC_LDS
  __builtin_amdgcn_global_load_async_to_lds_b32(
      (g_i32_t*)g_src, (l_i32_t*)lds_dst, 0, 0);
#else
  *lds_dst = *g_src;
#endif
}

__device__ __forceinline__ void wait_async_lds() {
#ifdef USE_ASYNC_LDS
#if defined(__has_builtin) && __has_builtin(__builtin_amdgcn_s_wait_asynccnt)
  __builtin_amdgcn_s_wait_asynccnt(0);
#else
  asm volatile("s_wait_asynccnt 0" ::: "memory");
#endif
#endif
}

// Stage a HB x HB halo tile of one field into this wave's LDS buffer.
// Zero outside the grid (reference zero-pads the derivative halo).
__device__ __forceinline__ void stage_tile_wave(float* lbuf, const float* gfld,
                                                int y0, int x0, int lane) {
  for (int e = lane; e < HB * HB; e += 32) {
    const int ry = e / HB;
    const int rx = e - ry * HB;
    const int hy = y0 - 2 + ry;
    const int hx = x0 - 2 + rx;
    float* l = lbuf + ry * LPAD + rx;
    if ((unsigned)hy < (unsigned)NYP && (unsigned)hx < (unsigned)NYP)
      async_copy_f32(l, gfld + hy * NYP + hx);
    else
      *l = 0.0f;
  }
}

// Banded stencil weight: halo offset d relative to output index.
// dbwd: f[i-2..i+1] -> offsets 0..3 with -C2,-C1,+C1,+C2 (all * 1/DX).
__device__ __forceinline__ float wcoef_bwd(int d) {
  if (d == 0) return -C2F * INVDX;
  if (d == 1) return -C1F * INVDX;
  if (d == 2) return  C1F * INVDX;
  if (d == 3) return  C2F * INVDX;
  return 0.0f;
}
// dfwd: f[i-1..i+2] -> offsets 1..4 with -C2,-C1,+C1,+C2.
__device__ __forceinline__ float wcoef_fwd(int d) { return wcoef_bwd(d - 1); }

// y-derivative of a staged tile: D(16x16) = W(16x20) * F(20x16).
// wv[c] is this lane's A-matrix fragment for K-chunk c (constant weights).
__device__ __forceinline__ v8f deriv_y(const float* lbuf, const v2f* wv,
                                       int lane, int fwd) {
#ifdef USE_WMMA
  const int n  = lane & 15;
  const int kb = (lane >> 4) << 1;    // 0 or 2
  v8f acc = {0.f, 0.f, 0.f, 0.f, 0.f, 0.f, 0.f, 0.f};
#pragma unroll
  for (int c = 0; c < 5; ++c) {
    v2f b;
    b.x = lbuf[(4 * c + kb) * LPAD + 2 + n];
    b.y = lbuf[(4 * c + kb + 1) * LPAD + 2 + n];
    acc = __builtin_amdgcn_wmma_f32_16x16x4_f32(
        false, wv[c], false, b, (short)0, acc, false, false);
  }
  return acc;
#else
  (void)wv;
  v8f r;
  const int n = lane & 15;
#pragma unroll
  for (int v = 0; v < 8; ++v) {
    const int m = v + ((lane >> 4) << 3);
    const int ci = (m + 2) * LPAD + n + 2;
    r[v] = fwd
      ? (C1F * (lbuf[ci + LPAD] - lbuf[ci]) +
         C2F * (lbuf[ci + 2 * LPAD] - lbuf[ci - LPAD])) * INVDX
      : (C1F * (lbuf[ci] - lbuf[ci - LPAD]) +
         C2F * (lbuf[ci + LPAD] - lbuf[ci - 2 * LPAD])) * INVDX;
  }
  return r;
#endif
}

// x-derivative: D(16x16) = F(16x20) * W(20x16); same per-lane weight
// fragment serves as the B-matrix by the m<->n symmetry of the band.
__device__ __forceinline__ v8f deriv_x(const float* lbuf, const v2f* wv,
                                       int lane, int fwd) {
#ifdef USE_WMMA
  const int m  = lane & 15;
  const int kb = (lane >> 4) << 1;
  v8f acc = {0.f, 0.f, 0.f, 0.f, 0.f, 0.f, 0.f, 0.f};
#pragma unroll
  for (int c = 0; c < 5; ++c) {
    v2f a;
    a.x = lbuf[(m + 2) * LPAD + 4 * c + kb];
    a.y = lbuf[(m + 2) * LPAD + 4 * c + kb + 1];
    acc = __builtin_amdgcn_wmma_f32_16x16x4_f32(
        false, a, false, wv[c], (short)0, acc, false, false);
  }
  return acc;
#else
  (void)wv;
  v8f r;
  const int n = lane & 15;
#pragma unroll
  for (int v = 0; v < 8; ++v) {
    const int m = v + ((lane >> 4) << 3);
    const int ci = (m + 2) * LPAD + n + 2;
    r[v] = fwd
      ? (C1F * (lbuf[ci + 1] - lbuf[ci]) +
         C2F * (lbuf[ci + 2] - lbuf[ci - 1])) * INVDX
      : (C1F * (lbuf[ci] - lbuf[ci - 1]) +
         C2F * (lbuf[ci + 1] - lbuf[ci - 2])) * INVDX;
  }
  return r;
#endif
}

// Device-wide barrier: monotonic counter (zeroed each launch); all NUMWG
// workgroups are resident (persistent launch), so the spin cannot deadlock.
__device__ __forceinline__ void gsync(unsigned* cnt, unsigned goal) {
  __threadfence();
  __syncthreads();
  if (threadIdx.x == 0) {
    __hip_atomic_fetch_add(cnt, 1u, __ATOMIC_RELEASE, __HIP_MEMORY_SCOPE_AGENT);
    while (__hip_atomic_load(cnt, __ATOMIC_ACQUIRE, __HIP_MEMORY_SCOPE_AGENT) < goal) {
      __builtin_amdgcn_s_sleep(2);
    }
  }
  __syncthreads();
  __threadfence();
}

// ---------------------------------------------------------------------------
// Setup kernels
// ---------------------------------------------------------------------------
__global__ void zero_ws(float* ws, int n) {
  for (int i = blockIdx.x * blockDim.x + threadIdx.x; i < n;
       i += gridDim.x * blockDim.x)
    ws[i] = 0.0f;
}

__global__ void prep_mats(const float* __restrict__ lamb,
                          const float* __restrict__ mu,
                          const float* __restrict__ buoy,
                          float* __restrict__ ws) {
  const int i = blockIdx.x * blockDim.x + threadIdx.x;
  if (i >= NN) return;
  const int y = i / NYP, x = i - (i / NYP) * NYP;
  const int iy = min(max(y - PMLW, 0), NYI - 1);
  const int ix = min(max(x - PMLW, 0), NYI - 1);
  const float l = lamb[iy * NYI + ix];
  const float m = mu[iy * NYI + ix];
  const float b = buoy[iy * NYI + ix];
  const float l2m = l + 2.0f * m;
  ws[O_LAM + i] = l;
  ws[O_MU  + i] = m;
  ws[O_BUO + i] = b;
  ws[O_L2M + i] = l2m;
  const float v = sqrtf(l2m * b);
  atomicMax((unsigned*)(ws + O_MAXV), __float_as_uint(v));  // v > 0
}

__global__ void prep_prof(float* __restrict__ ws) {
  const int i = threadIdx.x;
  if (i >= NYP) return;
  const float maxv = __uint_as_float(*(const unsigned*)(ws + O_MAXV));
  const float sig_max = 3.0f * maxv * 6.907755278982137f / (2.0f * PMLW * 4.0f);
  const float fi = (float)i;
  const float d = fmaxf(fmaxf((float)PMLW - fi, 0.0f),
                        fmaxf(fi - (float)(NYP - 1 - PMLW), 0.0f)) *
                  (1.0f / (float)PMLW);
  const float b = expf(-sig_max * d * d * DTF);
  ws[O_BY + i] = b;
  ws[O_AY + i] = b - 1.0f;
  ws[O_BX + i] = b;
  ws[O_AX + i] = b - 1.0f;
}

// ---------------------------------------------------------------------------
// Persistent time-stepping kernel (one 16x16 tile per wave)
// ---------------------------------------------------------------------------
__global__ __launch_bounds__(BLK) void elastic_fdtd_persistent(
    float* __restrict__ ws,
    const float* __restrict__ amps,    // [2,8,NTT]
    const int*   __restrict__ sloc,    // [2,8,2]
    const int*   __restrict__ rloc,    // [2,64,2]
    float*       __restrict__ out)     // [2,64,NTT]
{
  __shared__ float sbufs[WPB][3][LSZ];

  unsigned* cnt = (unsigned*)ws;
  const float* by = ws + O_BY;
  const float* ay = ws + O_AY;
  const float* bx = ws + O_BX;
  const float* ax = ws + O_AX;
  const float* lamP = ws + O_LAM;
  const float* muP  = ws + O_MU;
  const float* buoP = ws + O_BUO;
  const float* l2mP = ws + O_L2M;

  float* fld   = ws + O_FLD;
  float* vy    = fld + 0  * (NSH * NN);
  float* vx    = fld + 1  * (NSH * NN);
  float* syy   = fld + 2  * (NSH * NN);
  float* sxy   = fld + 3  * (NSH * NN);
  float* sxx   = fld + 4  * (NSH * NN);
  float* mvyy  = fld + 5  * (NSH * NN);
  float* mvyx  = fld + 6  * (NSH * NN);
  float* mvxy  = fld + 7  * (NSH * NN);
  float* mvxx  = fld + 8  * (NSH * NN);
  float* msyyy = fld + 9  * (NSH * NN);
  float* msxyy = fld + 10 * (NSH * NN);
  float* msxyx = fld + 11 * (NSH * NN);
  float* msxxx = fld + 12 * (NSH * NN);

  const int tid  = (int)threadIdx.x;
  const int wave = tid >> 5;
  const int lane = tid & 31;
  const int gw   = blockIdx.x * WPB + wave;   // global wave id
  float* bA = &sbufs[wave][0][0];
  float* bB = &sbufs[wave][1][0];
  float* bC = &sbufs[wave][2][0];

  // Per-lane constant stencil fragments (A for y-derivs / B for x-derivs).
  v2f wvb[5], wvf[5];
  {
    const int m  = lane & 15;
    const int kb = (lane >> 4) << 1;
#pragma unroll
    for (int c = 0; c < 5; ++c) {
      wvb[c].x = wcoef_bwd(4 * c + kb - m);
      wvb[c].y = wcoef_bwd(4 * c + kb + 1 - m);
      wvf[c].x = wcoef_fwd(4 * c + kb - m);
      wvf[c].y = wcoef_fwd(4 * c + kb + 1 - m);
    }
  }

  const int n    = lane & 15;           // output column within tile
  const int moff = (lane >> 4) << 3;    // +8 rows for upper half-wave
  unsigned seq = 0;

  for (int t = 0; t < NTT; ++t) {
    // ---------------- velocity half-step (+ source injection) -------------
    for (int tb = gw; tb < NTILES; tb += NWAVES) {
      const int sh = tb / (TPD * TPD);
      const int rr = tb - sh * (TPD * TPD);
      const int y0 = (rr / TPD) * TILE;
      const int x0 = (rr % TPD) * TILE;
      const int sb = sh * NN;

      stage_tile_wave(bA, syy + sb, y0, x0, lane);   // d/dy (bwd)
      stage_tile_wave(bB, sxy + sb, y0, x0, lane);   // d/dx and d/dy (bwd)
      stage_tile_wave(bC, sxx + sb, y0, x0, lane);   // d/dx (bwd)
      wait_async_lds();                               // per-wave ASYNCcnt

      const v8f d1 = deriv_y(bA, wvb, lane, 0);
      const v8f d2 = deriv_x(bB, wvb, lane, 0);
      const v8f d3 = deriv_y(bB, wvb, lane, 0);
      const v8f d4 = deriv_x(bC, wvb, lane, 0);

      const int x = x0 + n;
      if (x < NYP) {
        const float bxv = bx[x], axv = ax[x];
#pragma unroll
        for (int v = 0; v < 8; ++v) {
          const int y = y0 + v + moff;
          if (y < NYP) {
            const int pi = y * NYP + x;
            const int gi = sb + pi;
            const float byv = by[y], ayv = ay[y];

            float m1 = msyyy[gi]; m1 = byv * m1 + ayv * d1[v]; msyyy[gi] = m1;
            float m2 = msxyx[gi]; m2 = bxv * m2 + axv * d2[v]; msxyx[gi] = m2;
            const float b = buoP[pi];
            float vv = vy[gi] + DTF * b * ((d1[v] + m1) + (d2[v] + m2));
#pragma unroll
            for (int j = 0; j < NSRCS; ++j) {
              const int sy = sloc[(sh * NSRCS + j) * 2 + 0] + PMLW;
              const int sx = sloc[(sh * NSRCS + j) * 2 + 1] + PMLW;
              if (sy == y && sx == x)
                vv += DTF * b * amps[(sh * NSRCS + j) * NTT + t];
            }
            vy[gi] = vv;

            float m3 = msxyy[gi]; m3 = byv * m3 + ayv * d3[v]; msxyy[gi] = m3;
            float m4 = msxxx[gi]; m4 = bxv * m4 + axv * d4[v]; msxxx[gi] = m4;
            vx[gi] += DTF * b * ((d3[v] + m3) + (d4[v] + m4));
          }
        }
      }
    }
    ++seq; gsync(cnt, seq * NUMWG);

    // ---------------- record receivers (vy stable during stress phase) ----
    if (blockIdx.x == 0 && tid < NSH * NRECS) {
      const int sh = tid / NRECS, rr = tid % NRECS;
      const int ry = rloc[(sh * NRECS + rr) * 2 + 0] + PMLW;
      const int rx = rloc[(sh * NRECS + rr) * 2 + 1] + PMLW;
      out[(sh * NRECS + rr) * NTT + t] = vy[sh * NN + ry * NYP + rx];
    }

    // ---------------- stress half-step -------------------------------------
    for (int tb = gw; tb < NTILES; tb += NWAVES) {
      const int sh = tb / (TPD * TPD);
      const int rr = tb - sh * (TPD * TPD);
      const int y0 = (rr / TPD) * TILE;
      const int x0 = (rr % TPD) * TILE;
      const int sb = sh * NN;

      stage_tile_wave(bA, vy + sb, y0, x0, lane);    // d/dy, d/dx (fwd)
      stage_tile_wave(bB, vx + sb, y0, x0, lane);    // d/dx, d/dy (fwd)
      wait_async_lds();

      const v8f d1 = deriv_y(bA, wvf, lane, 1);   // dvy/dy
      const v8f d2 = deriv_x(bB, wvf, lane, 1);   // dvx/dx
      const v8f d3 = deriv_x(bA, wvf, lane, 1);   // dvy/dx
      const v8f d4 = deriv_y(bB, wvf, lane, 1);   // dvx/dy

      const int x = x0 + n;
      if (x < NYP) {
        const float bxv = bx[x], axv = ax[x];
#pragma unroll
        for (int v = 0; v < 8; ++v) {
          const int y = y0 + v + moff;
          if (y < NYP) {
            const int pi = y * NYP + x;
            const int gi = sb + pi;
            const float byv = by[y], ayv = ay[y];

            float m1 = mvyy[gi]; m1 = byv * m1 + ayv * d1[v]; mvyy[gi] = m1;
            float m2 = mvxx[gi]; m2 = bxv * m2 + axv * d2[v]; mvxx[gi] = m2;
            const float dvy_y = d1[v] + m1;
            const float dvx_x = d2[v] + m2;
            const float lam = lamP[pi], L = l2mP[pi];
            syy[gi] += DTF * (L * dvy_y + lam * dvx_x);
            sxx[gi] += DTF * (lam * dvy_y + L * dvx_x);

            float m3 = mvyx[gi]; m3 = bxv * m3 + axv * d3[v]; mvyx[gi] = m3;
            float m4 = mvxy[gi]; m4 = byv * m4 + ayv * d4[v]; mvxy[gi] = m4;
            sxy[gi] += DTF * muP[pi] * ((d3[v] + m3) + (d4[v] + m4));
          }
        }
      }
    }
    ++seq; gsync(cnt, seq * NUMWG);
  }
}

// ---------------------------------------------------------------------------
extern "C" void kernel_launch(void* const* d_in, const int* in_sizes, int n_in,
                              void* d_out, int out_size, void* d_ws, size_t ws_size,
                              hipStream_t stream) {
  (void)in_sizes; (void)n_in; (void)out_size; (void)ws_size;
  const float* lamb = (const float*)d_in[0];
  const float* mu   = (const float*)d_in[1];
  const float* buoy = (const float*)d_in[2];
  const float* amps = (const float*)d_in[3];
  const int*   sloc = (const int*)d_in[4];
  const int*   rloc = (const int*)d_in[5];
  float* out = (float*)d_out;
  float* ws  = (float*)d_ws;   // needs WS_FLOATS*4 ≈ 10.1 MiB of scratch

  zero_ws<<<2048, BLK, 0, stream>>>(ws, WS_FLOATS);
  prep_mats<<<(NN + BLK - 1) / BLK, BLK, 0, stream>>>(lamb, mu, buoy, ws);
  prep_prof<<<1, 512, 0, stream>>>(ws);
  elastic_fdtd_persistent<<<NUMWG, BLK, 0, stream>>>(ws, amps, sloc, rloc, out);
}